// PVTv2_Lawin_78340203479792
// MI455X (gfx1250) — compile-verified
//
#include <hip/hip_runtime.h>

// LawinHead forward for MI455X (gfx1250).  f32 end-to-end; GEMMs use
// V_WMMA_F32_16X16X4_F32.  Workspace need: 8,396,800 floats (~33.6 MB).

typedef __attribute__((ext_vector_type(2))) float v2f;
typedef __attribute__((ext_vector_type(8))) float v8f;

#define EPS 1e-5f

// ---------------- K1a: pooled = 0.5*(max+avg) over r x r (zero-padded) ----------------
__global__ void lawin_pool(const float* __restrict__ x, float* __restrict__ pooled,
                           int r, int pad) {
  int w = blockIdx.x;
  int wh = w >> 3, ww = w & 7;
  float invrr = 1.0f / (float)(r * r);
  for (int v = threadIdx.x; v < 512 * 64; v += blockDim.x) {
    int c = v >> 6, n = v & 63;
    int i = n >> 3, j = n & 7;
    int r0 = wh * 8 - pad + i * r;
    int c0 = ww * 8 - pad + j * r;
    float mx = -3.4028235e38f, sum = 0.0f;
    for (int a = 0; a < r; ++a) {
      int row = r0 + a;
      bool rin = (row >= 0) && (row < 64);
      for (int b = 0; b < r; ++b) {
        int col = c0 + b;
        float val = (rin && col >= 0 && col < 64) ? x[c * 4096 + row * 64 + col] : 0.0f;
        mx = fmaxf(mx, val);
        sum += val;
      }
    }
    pooled[w * 32768 + v] = 0.5f * (mx + sum * invrr);
  }
}

// ---------------- K1b: per-token LayerNorm stats over 512 channels ----------------
__global__ void lawin_stats(const float* __restrict__ pooled, float* __restrict__ stats) {
  int w = blockIdx.x;
  int n = threadIdx.x;  // 64 threads
  const float* p = pooled + w * 32768 + n;
  float s = 0.0f, ss = 0.0f;
  for (int c = 0; c < 512; ++c) {
    float v = p[c * 64];
    s += v; ss += v * v;
  }
  float m = s * (1.0f / 512.0f);
  float var = ss * (1.0f / 512.0f) - m * m;
  stats[w * 64 + n] = m;
  stats[4096 + w * 64 + n] = rsqrtf(var + EPS);
}

// ---------------- K1c: LN apply + per-head position mixing ----------------
// ctx[c][m] = ln[c][m] + pm_b[h][m] + sum_n ln[c][n]*pm_w[h][m][n]
__global__ void lawin_mix(const float* __restrict__ pooled, const float* __restrict__ stats,
                          const float* __restrict__ lng, const float* __restrict__ lnb,
                          const float* __restrict__ pmw, const float* __restrict__ pmb,
                          float* __restrict__ ctx, int cperh) {
  __shared__ float srow[8][64];
  __shared__ float smean[64];
  __shared__ float sinv[64];
  int w = blockIdx.x;
  int tid = threadIdx.x, lane = tid & 31, wid = tid >> 5;
  if (tid < 64) smean[tid] = stats[w * 64 + tid];
  else if (tid < 128) sinv[tid - 64] = stats[4096 + w * 64 + (tid - 64)];
  __syncthreads();
  for (int c = wid; c < 512; c += 8) {
    int head = c / cperh;
    float g = lng[c], b = lnb[c];
    const float* prow = pooled + w * 32768 + c * 64;
    srow[wid][lane]      = (prow[lane]      - smean[lane])      * sinv[lane]      * g + b;
    srow[wid][lane + 32] = (prow[lane + 32] - smean[lane + 32]) * sinv[lane + 32] * g + b;
    __syncthreads();  // uniform trip count across all warps
    const float* wrow = pmw + head * 4096;
    float* orow = ctx + w * 32768 + c * 64;
    for (int mi = 0; mi < 2; ++mi) {
      int m = lane + mi * 32;
      float acc = pmb[head * 64 + m] + srow[wid][m];
      const float* wm = wrow + m * 64;
#pragma unroll 8
      for (int n = 0; n < 64; ++n) acc += srow[wid][n] * wm[n];
      orow[m] = acc;
    }
    __syncthreads();
  }
}

// ---------------- K2: g/theta/phi projections, D[256,64] = W[256,512] @ X[512,64] ----------------
__global__ void lawin_proj(const float* __restrict__ x, const float* __restrict__ ctx,
                           const float* __restrict__ gw, const float* __restrict__ gb,
                           const float* __restrict__ tw, const float* __restrict__ tb,
                           const float* __restrict__ pw, const float* __restrict__ pb,
                           float* __restrict__ gout, float* __restrict__ tout,
                           float* __restrict__ pout) {
  __shared__ float sx[64][64];  // 16 KB K-chunk of X
  int w = blockIdx.x, which = blockIdx.y;
  const float* W; const float* bias; float* out;
  if (which == 0)      { W = gw; bias = gb; out = gout; }
  else if (which == 1) { W = tw; bias = tb; out = tout; }
  else                 { W = pw; bias = pb; out = pout; }
  bool qmode = (which == 1);  // theta consumes raw query window from x
  int wh = w >> 3, ww = w & 7;
  int tid = threadIdx.x, lane = tid & 31, wid = tid >> 5;
  int lrow = lane & 15;
  int khalf = (lane >> 4) * 2;

  v8f acc[8];
  for (int t = 0; t < 8; ++t)
    for (int e = 0; e < 8; ++e) acc[t][e] = 0.0f;

  for (int kc = 0; kc < 8; ++kc) {
    for (int v = tid; v < 4096; v += 256) {
      int kk = v >> 6, n = v & 63;
      int cch = kc * 64 + kk;
      float val;
      if (qmode) val = x[cch * 4096 + (wh * 8 + (n >> 3)) * 64 + (ww * 8 + (n & 7))];
      else       val = ctx[w * 32768 + cch * 64 + n];
      sx[kk][n] = val;
    }
    __syncthreads();
#pragma unroll 4
    for (int kb = 0; kb < 16; ++kb) {
      int kbase = kb * 4 + khalf;
      int kglob = kc * 64 + kbase;
      int m0 = (wid * 2 + 0) * 16 + lrow;
      int m1 = (wid * 2 + 1) * 16 + lrow;
      v2f a0, a1;
      a0.x = W[m0 * 512 + kglob]; a0.y = W[m0 * 512 + kglob + 1];
      a1.x = W[m1 * 512 + kglob]; a1.y = W[m1 * 512 + kglob + 1];
#pragma unroll
      for (int nt = 0; nt < 4; ++nt) {
        v2f bfrag;
        int n = nt * 16 + lrow;
        bfrag.x = sx[kbase][n];
        bfrag.y = sx[kbase + 1][n];
        acc[nt]     = __builtin_amdgcn_wmma_f32_16x16x4_f32(false, a0, false, bfrag,
                                                            (short)0, acc[nt], false, false);
        acc[4 + nt] = __builtin_amdgcn_wmma_f32_16x16x4_f32(false, a1, false, bfrag,
                                                            (short)0, acc[4 + nt], false, false);
      }
    }
    __syncthreads();
  }

  int nlo = lane & 15;
  int mhi = (lane >> 4) * 8;
  for (int mi = 0; mi < 2; ++mi) {
    int mbase = (wid * 2 + mi) * 16;
    for (int nt = 0; nt < 4; ++nt) {
      v8f a = acc[mi * 4 + nt];
      for (int v = 0; v < 8; ++v) {
        int m = mbase + v + mhi;
        int n = nt * 16 + nlo;
        out[w * 16384 + m * 64 + n] = a[v] + bias[m];
      }
    }
  }
}

// ---------------- K3: per-head attention (scores, softmax, y) ----------------
__global__ void lawin_attn(const float* __restrict__ thbuf, const float* __restrict__ phbuf,
                           const float* __restrict__ gbuf, float* __restrict__ ybuf, int d) {
  __shared__ float s[64][64];
  int w = blockIdx.x, h = blockIdx.y;
  int tid = threadIdx.x;
  const float* th = thbuf + w * 16384 + h * d * 64;
  const float* ph = phbuf + w * 16384 + h * d * 64;
  const float* g  = gbuf  + w * 16384 + h * d * 64;
  float scale = rsqrtf((float)d);
  for (int v = tid; v < 4096; v += 256) {
    int n = v >> 6, m = v & 63;
    float acc = 0.0f;
    for (int k = 0; k < d; ++k) acc += th[k * 64 + n] * ph[k * 64 + m];
    s[n][m] = acc * scale;
  }
  __syncthreads();
  if (tid < 64) {
    float mx = -3.4028235e38f;
    for (int m = 0; m < 64; ++m) mx = fmaxf(mx, s[tid][m]);
    float sum = 0.0f;
    for (int m = 0; m < 64; ++m) {
      float e = __expf(s[tid][m] - mx);
      s[tid][m] = e; sum += e;
    }
    float inv = 1.0f / sum;
    for (int m = 0; m < 64; ++m) s[tid][m] *= inv;
  }
  __syncthreads();
  for (int v = tid; v < d * 64; v += 256) {
    int dd = v >> 6, n = v & 63;
    float acc = 0.0f;
    for (int m = 0; m < 64; ++m) acc += s[n][m] * g[dd * 64 + m];
    ybuf[w * 16384 + (h * d + dd) * 64 + n] = acc;
  }
}

// ---------------- K4: conv_out[512,256] @ y[256,64] + BN + residual ----------------
__global__ void lawin_out(const float* __restrict__ x, const float* __restrict__ ybuf,
                          const float* __restrict__ cow,
                          const float* __restrict__ bng, const float* __restrict__ bnb,
                          const float* __restrict__ bnm, const float* __restrict__ bnv,
                          float* __restrict__ outp) {
  __shared__ float sx[64][64];
  int w = blockIdx.x, half = blockIdx.y;
  int wh = w >> 3, ww = w & 7;
  int tid = threadIdx.x, lane = tid & 31, wid = tid >> 5;
  int lrow = lane & 15;
  int khalf = (lane >> 4) * 2;

  v8f acc[8];
  for (int t = 0; t < 8; ++t)
    for (int e = 0; e < 8; ++e) acc[t][e] = 0.0f;

  const float* y = ybuf + w * 16384;
  for (int kc = 0; kc < 4; ++kc) {
    for (int v = tid; v < 4096; v += 256)
      sx[v >> 6][v & 63] = y[(kc * 64 + (v >> 6)) * 64 + (v & 63)];
    __syncthreads();
#pragma unroll 4
    for (int kb = 0; kb < 16; ++kb) {
      int kbase = kb * 4 + khalf;
      int kglob = kc * 64 + kbase;
      int m0 = half * 256 + (wid * 2) * 16 + lrow;
      int m1 = m0 + 16;
      v2f a0, a1;
      a0.x = cow[m0 * 256 + kglob]; a0.y = cow[m0 * 256 + kglob + 1];
      a1.x = cow[m1 * 256 + kglob]; a1.y = cow[m1 * 256 + kglob + 1];
#pragma unroll
      for (int nt = 0; nt < 4; ++nt) {
        v2f bfrag;
        int n = nt * 16 + lrow;
        bfrag.x = sx[kbase][n];
        bfrag.y = sx[kbase + 1][n];
        acc[nt]     = __builtin_amdgcn_wmma_f32_16x16x4_f32(false, a0, false, bfrag,
                                                            (short)0, acc[nt], false, false);
        acc[4 + nt] = __builtin_amdgcn_wmma_f32_16x16x4_f32(false, a1, false, bfrag,
                                                            (short)0, acc[4 + nt], false, false);
      }
    }
    __syncthreads();
  }

  int nlo = lane & 15;
  int mhi = (lane >> 4) * 8;
  for (int mi = 0; mi < 2; ++mi) {
    int mbase = half * 256 + (wid * 2 + mi) * 16;
    for (int nt = 0; nt < 4; ++nt) {
      v8f a = acc[mi * 4 + nt];
      for (int v = 0; v < 8; ++v) {
        int o = mbase + v + mhi;
        int n = nt * 16 + nlo;
        float z = a[v];
        z = (z - bnm[o]) * rsqrtf(bnv[o] + EPS) * bng[o] + bnb[o];
        int row = wh * 8 + (n >> 3), col = ww * 8 + (n & 7);
        outp[o * 4096 + row * 64 + col] = x[o * 4096 + row * 64 + col] + z;
      }
    }
  }
}

extern "C" void kernel_launch(void* const* d_in, const int* in_sizes, int n_in,
                              void* d_out, int out_size, void* d_ws, size_t ws_size,
                              hipStream_t stream) {
  // Input order: x first, then 45 param leaves (scales '2','4','8', keys alphabetized).
  int xi = 0, pstart = 1;
  if (in_sizes[0] != 512 * 64 * 64) { xi = n_in - 1; pstart = 0; }  // defensive: params-first flatten
  const float* x = (const float*)d_in[xi];
  float* out = (float*)d_out;
  float* ws = (float*)d_ws;

  float* pooled = ws;             // 2,097,152 f
  float* stats  = ws + 2097152;   //     8,192 f
  float* ctx    = ws + 2105344;   // 2,097,152 f
  float* gbuf   = ws + 4202496;   // 1,048,576 f
  float* thbuf  = ws + 5251072;   // 1,048,576 f
  float* phbuf  = ws + 6299648;   // 1,048,576 f
  float* ybuf   = ws + 7348224;   // 1,048,576 f  (end 8,396,800 f)

  struct Sc { int r, h, slot, outoff; };
  // output order r=8,4,2 ; param slots sorted '2'(0),'4'(1),'8'(2)
  const Sc sc[3] = { {8, 64, 2, 0}, {4, 16, 1, 512}, {2, 4, 0, 1024} };

  for (int s = 0; s < 3; ++s) {
    int r = sc[s].r, h = sc[s].h;
    int base = pstart + sc[s].slot * 15;
    const float* bn_b = (const float*)d_in[base + 0];
    const float* bn_g = (const float*)d_in[base + 1];
    const float* bn_m = (const float*)d_in[base + 2];
    const float* bn_v = (const float*)d_in[base + 3];
    const float* co_w = (const float*)d_in[base + 4];
    const float* g_b  = (const float*)d_in[base + 5];
    const float* g_w  = (const float*)d_in[base + 6];
    const float* ln_b = (const float*)d_in[base + 7];
    const float* ln_g = (const float*)d_in[base + 8];
    const float* pm_b = (const float*)d_in[base + 9];
    const float* pm_w = (const float*)d_in[base + 10];
    const float* ph_b = (const float*)d_in[base + 11];
    const float* ph_w = (const float*)d_in[base + 12];
    const float* th_b = (const float*)d_in[base + 13];
    const float* th_w = (const float*)d_in[base + 14];
    int pad = (r - 1) * 4;

    lawin_pool<<<64, 256, 0, stream>>>(x, pooled, r, pad);
    lawin_stats<<<64, 64, 0, stream>>>(pooled, stats);
    lawin_mix<<<64, 256, 0, stream>>>(pooled, stats, ln_g, ln_b, pm_w, pm_b, ctx, 512 / h);
    lawin_proj<<<dim3(64, 3), 256, 0, stream>>>(x, ctx, g_w, g_b, th_w, th_b, ph_w, ph_b,
                                                gbuf, thbuf, phbuf);
    lawin_attn<<<dim3(64, h), 256, 0, stream>>>(thbuf, phbuf, gbuf, ybuf, 256 / h);
    lawin_out<<<dim3(64, 2), 256, 0, stream>>>(x, ybuf, co_w, bn_g, bn_b, bn_m, bn_v,
                                               out + sc[s].outoff * 4096);
  }
}